// GNN_84189948936590
// MI455X (gfx1250) — compile-verified
//
#include <hip/hip_runtime.h>
#include <hip/hip_bf16.h>
#include <stdint.h>

#define N_NODES 100000
#define N_EDGES 3200000

typedef __attribute__((ext_vector_type(2)))  float     v2f;
typedef __attribute__((ext_vector_type(8)))  float     v8f;
typedef __attribute__((ext_vector_type(2)))  long long v2l;

// Workspace layout (floats):
//   combos[0   .. 1023] : B matrix, 16 columns x 64 rows, column-major
//                         (col 0 = u, col 1 = v+w, col 2 = z, cols 3..15 = 0)
//   combos[1024]        : c1  = W_rel2 . b_rel1
//   combos[1025]        : c23 = b_rel2 + W_root2 . b_rel1
//   p[N], q[N], acc[N] follow.

// -------- kernel 0: collapse weights into rank-1 combos (trivial cost) -----
__global__ void gnn_prep(const float* __restrict__ Wr1, const float* __restrict__ br1,
                         const float* __restrict__ Wo1, const float* __restrict__ Wr2,
                         const float* __restrict__ br2, const float* __restrict__ Wo2,
                         float* __restrict__ combos) {
    int t = threadIdx.x;
    if (t < 64) {
        float u = 0.f, vw = 0.f, z = 0.f;
        for (int j = 0; j < 128; ++j) {
            float a2 = Wr2[j], b2 = Wo2[j];
            float r1 = Wr1[j * 64 + t], o1 = Wo1[j * 64 + t];
            u  += a2 * r1;
            vw += a2 * o1 + b2 * r1;
            z  += b2 * o1;
        }
        combos[0 * 64 + t] = u;
        combos[1 * 64 + t] = vw;
        combos[2 * 64 + t] = z;
        for (int col = 3; col < 16; ++col) combos[col * 64 + t] = 0.0f;  // real zeros
    } else if (t == 64) {
        float c1 = 0.f, c3 = 0.f;
        for (int j = 0; j < 128; ++j) { c1 += Wr2[j] * br1[j]; c3 += Wo2[j] * br1[j]; }
        combos[1024] = c1;
        combos[1025] = br2[0] + c3;
    }
}

// -------- kernel 1: per-16-node tile, WMMA [16x64]x[64x16] ------------------
// Result column 0 -> p[node], column 1 -> q[node], column 2 -> out[node]=t+c23.
__global__ void gnn_node_wmma(const float* __restrict__ x,
                              const float* __restrict__ combos,
                              float* __restrict__ p, float* __restrict__ q,
                              float* __restrict__ acc, float* __restrict__ out) {
    const int lane  = threadIdx.x & 31;
    const int wave  = threadIdx.x >> 5;
    const int tile  = blockIdx.x * (blockDim.x >> 5) + wave;
    const int nTile = N_NODES / 16;          // 6250 exactly
    if (tile >= nTile) return;               // uniform per wave: EXEC stays all-1s
    const int node0 = tile * 16;
    const int m = lane & 15;                 // A row (node) / B,C column
    const int h = lane >> 4;                 // lane half
    const v2f* __restrict__ x2 = (const v2f*)(x + (size_t)(node0 + m) * 64);
    const v2f* __restrict__ b2 = (const v2f*)(combos + m * 64);   // padded: valid for all m
    const float c23 = combos[1025];

    // Preload loop-invariant B fragments (32 floats/lane, branch-free b64 loads)
    v2f bfrag[16];
#pragma unroll
    for (int t = 0; t < 16; ++t) bfrag[t] = b2[2 * t + h];

    // A 16x4 f32 layout: lanes 0-15 hold K=k0,k0+1; lanes 16-31 K=k0+2,k0+3
    v8f c = {};
#pragma unroll
    for (int t = 0; t < 16; ++t) {
        v2f av = x2[2 * t + h];
        c = __builtin_amdgcn_wmma_f32_16x16x4_f32(false, av, false, bfrag[t],
                                                  (short)0, c, false, false);
    }

    // re-zero the scatter accumulator every call (ws is not re-poisoned)
    if (h == 0) acc[node0 + m] = 0.0f;

    // C/D layout: VGPR i -> row (i + 8*h), column = m
#pragma unroll
    for (int i = 0; i < 8; ++i) {
        int row = node0 + i + 8 * h;
        float val = c[i];
        if (m == 0)      p[row] = val;
        else if (m == 1) q[row] = val;
        else if (m == 2) out[row] = val + c23;   // t + c2 + c3
    }
}

// -------- edge scatter: acc[dst] += val[src], 2 edges/thread ----------------
// Edge list is streamed once (no reuse): nontemporal loads keep it from
// evicting the L2-resident node vectors.
__global__ void gnn_scatter(const long long* __restrict__ ei,
                            const float* __restrict__ val,
                            float* __restrict__ acc) {
    int e2 = blockIdx.x * blockDim.x + threadIdx.x;
    if (e2 >= N_EDGES / 2) return;
    const v2l* s2 = (const v2l*)ei;                    // src row
    const v2l* d2 = (const v2l*)(ei + (size_t)N_EDGES); // dst row
    v2l s = __builtin_nontemporal_load(&s2[e2]);
    v2l d = __builtin_nontemporal_load(&d2[e2]);
    atomicAdd(&acc[(int)d.x], val[(int)s.x]);
    atomicAdd(&acc[(int)d.y], val[(int)s.y]);
}

// -------- combine: q[n] = acc[n] + q[n] + c1 (folds deg*c1 into 2nd scatter)
__global__ void gnn_combine(const float* __restrict__ acc, float* __restrict__ q,
                            const float* __restrict__ combos) {
    int n = blockIdx.x * blockDim.x + threadIdx.x;
    if (n >= N_NODES) return;
    q[n] = acc[n] + q[n] + combos[1024];
}

extern "C" void kernel_launch(void* const* d_in, const int* in_sizes, int n_in,
                              void* d_out, int out_size, void* d_ws, size_t ws_size,
                              hipStream_t stream) {
    const float*     x   = (const float*)d_in[0];
    const long long* ei  = (const long long*)d_in[1];   // int64 [2, E]
    // d_in[2] = batch (unused)
    const float*     Wr1 = (const float*)d_in[3];
    const float*     br1 = (const float*)d_in[4];
    const float*     Wo1 = (const float*)d_in[5];
    const float*     Wr2 = (const float*)d_in[6];
    const float*     br2 = (const float*)d_in[7];
    const float*     Wo2 = (const float*)d_in[8];
    float* out = (float*)d_out;

    float* ws     = (float*)d_ws;
    float* combos = ws;                       // 1024 + 2 floats (padded B + scalars)
    float* p      = ws + 1088;                // [N]
    float* q      = p + N_NODES;              // [N]
    float* acc    = q + N_NODES;              // [N]  (~1.2 MB total)

    gnn_prep<<<1, 128, 0, stream>>>(Wr1, br1, Wo1, Wr2, br2, Wo2, combos);

    const int nTile  = N_NODES / 16;                       // 6250
    const int wpb    = 8;                                  // 256 threads = 8 waves
    const int blocks = (nTile + wpb - 1) / wpb;            // 782
    gnn_node_wmma<<<blocks, 256, 0, stream>>>(x, combos, p, q, acc, out);

    const int eblocks = (N_EDGES / 2 + 255) / 256;
    gnn_scatter<<<eblocks, 256, 0, stream>>>(ei, p, acc);            // acc = S p
    gnn_combine<<<(N_NODES + 255) / 256, 256, 0, stream>>>(acc, q, combos);
    gnn_scatter<<<eblocks, 256, 0, stream>>>(ei, q, out);            // out += S(...)
}